// ModelNew_73315091744718
// MI455X (gfx1250) — compile-verified
//
#include <hip/hip_runtime.h>

// Problem constants (match reference)
#define BATCH 4096
#define IN    8192   // INPUT_SIZE
#define HID   8192   // HIDDEN_SIZE
#define NCHUNK 8     // row chunks for phase-1 parallelism

typedef __attribute__((ext_vector_type(2))) float v2f;
typedef __attribute__((ext_vector_type(8))) float v8f;

// ---------------------------------------------------------------------------
// Phase 1: wsum_partial[chunk][i] = sum over rows k in chunk of W[k][i]
// Computed as ones(16x4) @ W(4x16) tiles via V_WMMA_F32_16X16X4_F32.
// A is all-ones -> D[m][n] = sum_k B[k][n], identical for every m; we read
// row M=0 (acc[0], lanes 0..15 per ISA 16x16 f32 C/D layout).
//
// B fragment layout (32-bit 4x16, wave32): VGPR0 = K=0 (lanes 0-15) /
// K=2 (lanes 16-31); VGPR1 = K=1 / K=3. So per step lanes 0-15 load a
// contiguous 64B segment of row k, lanes 16-31 of row k+2 (coalesced).
// One wave per block so EXEC is all-1s at the WMMA (ISA requirement).
// ---------------------------------------------------------------------------
__global__ void __launch_bounds__(32)
wsum_partial_wmma(const float* __restrict__ W, float* __restrict__ partial) {
    const int lane  = threadIdx.x;          // 0..31
    const int strip = blockIdx.x;           // 0..IN/16-1  -> columns n0..n0+15
    const int chunk = blockIdx.y;           // 0..NCHUNK-1 -> rows
    const int n0    = strip * 16;
    const int kspan = HID / NCHUNK;         // 1024 rows per chunk
    const int kbeg  = chunk * kspan;
    const int kend  = kbeg + kspan;

    const int ncol = lane & 15;
    const int krow = (lane < 16) ? 0 : 2;   // which K-row this half-wave holds
    const float* colbase = W + (size_t)ncol + (size_t)n0;

    v2f a; a[0] = 1.0f; a[1] = 1.0f;        // A = ones (16x4)
    v8f acc = {};

    #pragma unroll 4
    for (int k = kbeg; k < kend; k += 4) {
        v2f b;
        b[0] = colbase[(size_t)(k + krow)     * (size_t)IN];  // K = k+krow
        b[1] = colbase[(size_t)(k + krow + 1) * (size_t)IN];  // K = k+krow+1
        // 8 args: (neg_a, A, neg_b, B, c_mod, C, reuse_a, reuse_b)
        acc = __builtin_amdgcn_wmma_f32_16x16x4_f32(
            false, a, false, b, (short)0, acc, false, false);
    }

    // D row M=0: acc[0] on lanes 0..15 holds columns n0..n0+15
    if (lane < 16) {
        partial[(size_t)chunk * IN + n0 + lane] = acc[0];
    }
}

// ---------------------------------------------------------------------------
// Phase 1b: wsum[i] = sum over chunks of partial[chunk][i]   (tiny: 256 KB)
// ---------------------------------------------------------------------------
__global__ void __launch_bounds__(256)
wsum_reduce(const float* __restrict__ partial, float* __restrict__ wsum) {
    const int i = blockIdx.x * blockDim.x + threadIdx.x;  // 0..IN-1
    float s = 0.0f;
    #pragma unroll
    for (int c = 0; c < NCHUNK; ++c) s += partial[(size_t)c * IN + i];
    wsum[i] = s;
}

// ---------------------------------------------------------------------------
// Phase 2: out[b] = 0.75 * dot(x[b,:], wsum)
// 4096 blocks x 256 threads; float4 streaming loads (8 iters/thread),
// LDS tree reduction. wsum (32 KB) stays resident in L2 across all blocks.
// ---------------------------------------------------------------------------
__global__ void __launch_bounds__(256)
rowdot(const float* __restrict__ x, const float* __restrict__ wsum,
       float* __restrict__ out) {
    const int b = blockIdx.x;
    const int t = threadIdx.x;

    const float4* __restrict__ xr = (const float4*)(x + (size_t)b * IN);
    const float4* __restrict__ ws = (const float4*)wsum;

    float s = 0.0f;
    #pragma unroll
    for (int i = t; i < IN / 4; i += 256) {
        __builtin_prefetch(&xr[i + 512], 0, 0);   // -> global_prefetch_b8
        float4 xv = xr[i];
        float4 wv = ws[i];
        s = fmaf(xv.x, wv.x, s);
        s = fmaf(xv.y, wv.y, s);
        s = fmaf(xv.z, wv.z, s);
        s = fmaf(xv.w, wv.w, s);
    }

    __shared__ float red[256];
    red[t] = s;
    __syncthreads();
    #pragma unroll
    for (int off = 128; off >= 1; off >>= 1) {
        if (t < off) red[t] += red[t + off];
        __syncthreads();
    }
    if (t == 0) out[b] = 0.75f * red[0];
}

// ---------------------------------------------------------------------------
extern "C" void kernel_launch(void* const* d_in, const int* in_sizes, int n_in,
                              void* d_out, int out_size, void* d_ws, size_t ws_size,
                              hipStream_t stream) {
    const float* x = (const float*)d_in[0];   // [BATCH, IN]
    const float* W = (const float*)d_in[1];   // [HID, IN]
    float* out = (float*)d_out;               // [BATCH]

    float* partial = (float*)d_ws;                       // NCHUNK * IN floats
    float* wsum    = partial + (size_t)NCHUNK * IN;      // IN floats

    dim3 g1(IN / 16, NCHUNK);                 // 512 strips x 8 chunks, 1 wave each
    wsum_partial_wmma<<<g1, 32, 0, stream>>>(W, partial);
    wsum_reduce<<<IN / 256, 256, 0, stream>>>(partial, wsum);
    rowdot<<<BATCH, 256, 0, stream>>>(x, wsum, out);
}